// ThirdOrderPooling_59837484368486
// MI455X (gfx1250) — compile-verified
//
#include <hip/hip_runtime.h>

typedef __attribute__((ext_vector_type(2))) float v2f;
typedef __attribute__((ext_vector_type(8))) float v8f;

constexpr int Dc     = 64;     // channel dim
constexpr int Nn     = 8192;   // reduction dim
constexpr int Bb     = 32;     // batch
constexpr int TILE_N = 128;    // rows staged in LDS per iteration
constexpr int LDSW   = 72;     // padded LDS row stride (rows k and k+2 hit disjoint banks)

// ---------------- Kernel 1: per-(b,c) mean over N ----------------
__global__ __launch_bounds__(256) void mean_kernel(const float* __restrict__ x,
                                                   float* __restrict__ mean) {
    const int b = blockIdx.x;
    const int t = threadIdx.x;
    const int c = t & 63;       // channel
    const int g = t >> 6;       // 0..3 n-group
    const float* xb = x + (size_t)b * Nn * Dc;
    float s = 0.0f;
    for (int n = g; n < Nn; n += 4)
        s += xb[(size_t)n * Dc + c];
    __shared__ float red[256];
    red[t] = s;
    __syncthreads();
    if (t < 64) {
        float m = red[t] + red[t + 64] + red[t + 128] + red[t + 192];
        mean[b * Dc + t] = m * (1.0f / (float)Nn);
    }
}

// ---------------- Kernel 2: third moment via fp32 WMMA ----------------
// One workgroup per (b, group of 4 d's). 16 waves: wave w handles
// d = dbase + (w>>2), f-block = (w&3), and ALL FOUR e-blocks (4 accumulators).
// Identity used: (x_d * x_e) * x_f = x_e * (x_d * x_f), so the d-weight is
// folded into the B operand (shared across the 4 e-tiles) and A is a raw
// x_e load — 6 ds_load_2addr + 2 v_mul per 4 WMMAs.
__global__ __launch_bounds__(512, 1) void third_kernel(const float* __restrict__ x,
                                                       const float* __restrict__ mean,
                                                       float* __restrict__ out) {
    const int b     = blockIdx.x >> 4;
    const int dbase = (blockIdx.x & 15) << 2;

    const float* xb = x + (size_t)b * Nn * Dc;
    const float* mb = mean + b * Dc;

    __shared__ float lds[TILE_N * LDSW];

    const int t     = threadIdx.x;
    const int lane  = t & 31;        // wave32
    const int w     = t >> 5;        // wave id 0..15
    const int j     = lane & 15;     // column within tile
    const int half  = lane >> 4;     // 0: K={0,1}, 1: K={2,3}
    const int d     = dbase + (w >> 2);
    const int fbase = (w & 3) << 4;

    // Cooperative staging: each of 512 threads loads 16 contiguous floats.
    const int lr = t >> 2;           // row 0..127
    const int lc = (t & 3) << 4;     // col start 0,16,32,48

    float4 m[4];
#pragma unroll
    for (int i = 0; i < 4; ++i) m[i] = *(const float4*)(mb + lc + 4 * i);

    v8f acc0 = {}, acc1 = {}, acc2 = {}, acc3 = {};

    for (int n0 = 0; n0 < Nn; n0 += TILE_N) {
        __syncthreads();   // previous compute done before LDS overwrite
        const float* src = xb + (size_t)(n0 + lr) * Dc + lc;
        float*       dst = &lds[lr * LDSW + lc];
#pragma unroll
        for (int i = 0; i < 4; ++i) {
            float4 v = *(const float4*)(src + 4 * i);
            v.x -= m[i].x; v.y -= m[i].y; v.z -= m[i].z; v.w -= m[i].w;
            *(float4*)(dst + 4 * i) = v;   // 288B row stride -> 16B aligned
        }
        __syncthreads();

#pragma unroll 4
        for (int k = 0; k < TILE_N; k += 4) {
            // Lane-half selects K sub-pair per ISA 16x4 fp32 A/B layout.
            const float* r0 = &lds[(k + 2 * half) * LDSW];
            const float* r1 = r0 + LDSW;
            const float xd0 = r0[d];           // uniform broadcast read
            const float xd1 = r1[d];
            v2f bv = { xd0 * r0[fbase + j], xd1 * r1[fbase + j] };
            v2f a0 = { r0[j],      r1[j]      };
            v2f a1 = { r0[16 + j], r1[16 + j] };
            v2f a2 = { r0[32 + j], r1[32 + j] };
            v2f a3 = { r0[48 + j], r1[48 + j] };
            acc0 = __builtin_amdgcn_wmma_f32_16x16x4_f32(false, a0, false, bv, (short)0, acc0, false, false);
            acc1 = __builtin_amdgcn_wmma_f32_16x16x4_f32(false, a1, false, bv, (short)0, acc1, false, false);
            acc2 = __builtin_amdgcn_wmma_f32_16x16x4_f32(false, a2, false, bv, (short)0, acc2, false, false);
            acc3 = __builtin_amdgcn_wmma_f32_16x16x4_f32(false, a3, false, bv, (short)0, acc3, false, false);
        }
    }

    // C/D layout: VGPR i, lane L -> row M = i + 8*(L>>4), col = L&15.
    const float scale = 1.0f / (float)Nn;
    const size_t base = (((size_t)b * Dc + d) << 12) + (size_t)fbase + j;
    const v8f accs[4] = { acc0, acc1, acc2, acc3 };
#pragma unroll
    for (int eb = 0; eb < 4; ++eb) {
#pragma unroll
        for (int i = 0; i < 8; ++i) {
            int e = (eb << 4) + i + 8 * half;
            out[base + ((size_t)e << 6)] = accs[eb][i] * scale;
        }
    }
}

extern "C" void kernel_launch(void* const* d_in, const int* in_sizes, int n_in,
                              void* d_out, int out_size, void* d_ws, size_t ws_size,
                              hipStream_t stream) {
    const float* x   = (const float*)d_in[0];
    float*       out = (float*)d_out;
    float*       mn  = (float*)d_ws;   // 32*64 floats = 8 KB

    mean_kernel<<<Bb, 256, 0, stream>>>(x, mn);
    third_kernel<<<Bb * 16, 512, 0, stream>>>(x, mn, out);
}